// GNNLSTMModel_15650860827371
// MI455X (gfx1250) — compile-verified
//
#include <hip/hip_runtime.h>
#include <cstdint>
#include <cstddef>

#define N_NODES 32768
#define N_EDGES 524288
#define HID 256
#define N_GRAPHS 64
#define N_CLASSES 4
#define NODES_PER_GRAPH (N_NODES / N_GRAPHS)
#define BN_EPS 1e-5f

#define KDIM 256            // K is 256 in both GEMMs (N_FEAT == HID == 256)
#define APAD 4              // LDS row pad: stride 260 -> conflict-free ds_load_b64
#define AROW (KDIM + APAD)

typedef float v2f __attribute__((ext_vector_type(2)));
typedef float v8f __attribute__((ext_vector_type(8)));

// ---------------- utility ----------------
__global__ __launch_bounds__(256) void k_zero(float* __restrict__ p, size_t n) {
  size_t i = (size_t)blockIdx.x * blockDim.x + threadIdx.x;
  size_t stride = (size_t)gridDim.x * blockDim.x;
  for (; i < n; i += stride) p[i] = 0.0f;
}

// ---------------- degree / norm coefficients ----------------
__global__ __launch_bounds__(256) void k_deg(const int* __restrict__ dst,
                                             float* __restrict__ deg) {
  int e = blockIdx.x * blockDim.x + threadIdx.x;
  if (e < N_EDGES) atomicAdd(&deg[dst[e]], 1.0f);
}

__global__ __launch_bounds__(256) void k_dinv(const float* __restrict__ degRaw,
                                              float* __restrict__ dinv,
                                              float* __restrict__ invdeg) {
  int i = blockIdx.x * blockDim.x + threadIdx.x;
  if (i < N_NODES) {
    float d = degRaw[i] + 1.0f;     // in-degree + self loop
    dinv[i] = rsqrtf(d);
    invdeg[i] = 1.0f / d;
  }
}

// ---------------- fp32 WMMA GEMM, K fixed at 256 ----------------
// C[M,N] = A[M,256] @ B[256,N], all row-major, N % 256 == 0, M % 16 == 0.
// Block (256 thr = 8 waves): 16-row strip (blockIdx.x) x 256-col panel (blockIdx.y).
// A strip (16x256 fp32, 16 KB) is staged in LDS once (coalesced float4 loads),
// then each wave computes TWO 16x16 tiles feeding V_WMMA_F32_16X16X4_F32 from
// ds_load_b64 (padded rows: stride 260 words -> all 64 banks hit exactly once).
// Per ISA 7.12.2:
//   A 16x4 f32 : lanes 0-15 K={k,k+1}, lanes 16-31 K={k+2,k+3}
//   B 4x16     : VGPR j, half h -> row k+2h+j, col lane%16
//   D 16x16    : VGPR r, half h -> row r+8h,   col lane%16
__global__ __launch_bounds__(256) void k_wmma_gemm256(const float* __restrict__ A,
                                                      const float* __restrict__ B,
                                                      float* __restrict__ C,
                                                      int N) {
  __shared__ float As[16 * AROW];
  const int tid = threadIdx.x;
  const int m0 = blockIdx.x << 4;
  const int n0 = blockIdx.y << 8;

  // cooperative, fully-coalesced load of the A strip into padded LDS
  {
    const float4* Ag = (const float4*)(A + (size_t)m0 * KDIM);
    #pragma unroll
    for (int i = 0; i < 4; ++i) {
      int linear = tid + (i << 8);     // float4 index within 16 x 64 grid
      int row = linear >> 6;
      int c4 = linear & 63;
      float4 v = Ag[(size_t)(row << 6) + c4];
      *(float4*)&As[row * AROW + (c4 << 2)] = v;
    }
  }
  __syncthreads();

  const int lane = tid & 31;
  const int wid = tid >> 5;
  const int lr = lane & 15;
  const int hi = lane >> 4;
  const int tn = n0 + (wid << 5);      // this wave owns cols [tn, tn+32)

  v8f acc0 = {}, acc1 = {};
  const float* ap = &As[lr * AROW + (hi << 1)];
  const float* bp = B + (size_t)(hi << 1) * N + tn + lr;
  const size_t bstep = (size_t)4 * N;

  #pragma unroll 4
  for (int k = 0; k < KDIM; k += 4) {
    v2f a;
    a.x = ap[k];
    a.y = ap[k + 1];                   // ds_load_b64, immediate offset
    v2f b0, b1;
    b0.x = bp[0];
    b0.y = bp[N];
    b1.x = bp[16];
    b1.y = bp[N + 16];
    acc0 = __builtin_amdgcn_wmma_f32_16x16x4_f32(false, a, false, b0,
                                                 (short)0, acc0, false, false);
    acc1 = __builtin_amdgcn_wmma_f32_16x16x4_f32(false, a, false, b1,
                                                 (short)0, acc1, false, false);
    bp += bstep;                        // single pointer bump, no per-iter MULs
  }

  float* crow = C + (size_t)(m0 + (hi << 3)) * N + tn + lr;
  #pragma unroll
  for (int r = 0; r < 8; ++r) {
    crow[0]  = acc0[r];
    crow[16] = acc1[r];
    crow += N;
  }
}

// ---------------- edge gather + scale + scatter-add ----------------
// 64 consecutive lanes per edge (float4 chunks) -> coalesced gather, f32 atomics scatter.
__global__ __launch_bounds__(256) void k_scatter(const float* __restrict__ h,
                                                 const int* __restrict__ src,
                                                 const int* __restrict__ dst,
                                                 const float* __restrict__ dinv,
                                                 float* __restrict__ agg) {
  long long gid = (long long)blockIdx.x * blockDim.x + threadIdx.x;
  int e = (int)(gid >> 6);
  if (e >= N_EDGES) return;
  int c = ((int)(gid & 63)) << 2;
  int s = src[e], d = dst[e];
  float coef = dinv[s] * dinv[d];
  const float4 v = *(const float4*)(h + (size_t)s * HID + c);
  float* ap = agg + (size_t)d * HID + c;
  atomicAdd(ap + 0, v.x * coef);
  atomicAdd(ap + 1, v.y * coef);
  atomicAdd(ap + 2, v.z * coef);
  atomicAdd(ap + 3, v.w * coef);
}

// ---------------- add self-loop term + bias, accumulate BN column stats ----------------
#define FIN_ROWS 128
__global__ __launch_bounds__(256) void k_finalize(float* __restrict__ y,   // agg, in/out
                                                  const float* __restrict__ h,
                                                  const float* __restrict__ invdeg,
                                                  const float* __restrict__ bias,
                                                  float* __restrict__ colsum,
                                                  float* __restrict__ colsq) {
  int c = threadIdx.x;                      // one column per thread (HID==256)
  int r0 = blockIdx.x * FIN_ROWS;
  float b = bias[c];
  float s = 0.f, sq = 0.f;
  for (int r = r0; r < r0 + FIN_ROWS; ++r) {
    size_t idx = (size_t)r * HID + c;
    float v = y[idx] + h[idx] * invdeg[r] + b;
    y[idx] = v;
    s += v;
    sq += v * v;
  }
  atomicAdd(&colsum[c], s);
  atomicAdd(&colsq[c], sq);
}

__global__ __launch_bounds__(256) void k_bnfin(const float* __restrict__ colsum,
                                               const float* __restrict__ colsq,
                                               const float* __restrict__ gamma,
                                               const float* __restrict__ beta,
                                               float* __restrict__ scale,
                                               float* __restrict__ shift) {
  int c = threadIdx.x;
  const float invN = 1.0f / (float)N_NODES;
  float mean = colsum[c] * invN;
  float var = colsq[c] * invN - mean * mean;   // biased variance
  float rstd = rsqrtf(var + BN_EPS);
  float sc = gamma[c] * rstd;
  scale[c] = sc;
  shift[c] = beta[c] - mean * sc;
}

__global__ __launch_bounds__(256) void k_bnrelu(const float* __restrict__ y,
                                                const float* __restrict__ scale,
                                                const float* __restrict__ shift,
                                                float* __restrict__ out) {
  size_t i = (size_t)blockIdx.x * blockDim.x + threadIdx.x;
  int c = (int)(i & (HID - 1));
  float v = y[i] * scale[c] + shift[c];
  out[i] = v > 0.f ? v : 0.f;
}

// ---------------- global mean pool (contiguous segments of 512 nodes) ----------------
__global__ __launch_bounds__(256) void k_pool(const float* __restrict__ h,
                                              float* __restrict__ pooled) {
  int g = blockIdx.x, c = threadIdx.x;
  const float* p = h + (size_t)g * NODES_PER_GRAPH * HID + c;
  float s = 0.f;
  for (int r = 0; r < NODES_PER_GRAPH; ++r) s += p[(size_t)r * HID];
  pooled[g * HID + c] = s * (1.0f / NODES_PER_GRAPH);
}

// ---------------- LSTM cell (h0 = c0 = 0) ----------------
__device__ __forceinline__ float sigmoidf_(float x) { return 1.0f / (1.0f + expf(-x)); }

__global__ __launch_bounds__(256) void k_lstm(const float* __restrict__ gates,
                                              const float* __restrict__ b_ih,
                                              const float* __restrict__ b_hh,
                                              float* __restrict__ h1,
                                              float* __restrict__ c1) {
  int idx = blockIdx.x * blockDim.x + threadIdx.x;   // N_GRAPHS * HID
  int g = idx >> 8;
  int c = idx & (HID - 1);
  const float* row = gates + (size_t)g * (4 * HID);
  float iv = row[c]           + b_ih[c]           + b_hh[c];
  float gv = row[c + 2 * HID] + b_ih[c + 2 * HID] + b_hh[c + 2 * HID];
  float ov = row[c + 3 * HID] + b_ih[c + 3 * HID] + b_hh[c + 3 * HID];
  // c1 = sigmoid(f)*c0 + sigmoid(i)*tanh(g), c0 == 0
  float cc = sigmoidf_(iv) * tanhf(gv);
  float hh = sigmoidf_(ov) * tanhf(cc);
  c1[idx] = cc;
  h1[idx] = hh;
}

// ---------------- FC + log_softmax (64 graphs x 4 classes, tiny) ----------------
__global__ __launch_bounds__(64) void k_fc(const float* __restrict__ h1,
                                           const float* __restrict__ Wfc,
                                           const float* __restrict__ bfc,
                                           float* __restrict__ logp) {
  int g = threadIdx.x;
  if (g >= N_GRAPHS) return;
  float l[N_CLASSES];
  #pragma unroll
  for (int c = 0; c < N_CLASSES; ++c) l[c] = bfc[c];
  const float* hr = h1 + (size_t)g * HID;
  for (int k = 0; k < HID; ++k) {
    float hv = hr[k];
    const float* wr = Wfc + (size_t)k * N_CLASSES;
    #pragma unroll
    for (int c = 0; c < N_CLASSES; ++c) l[c] += hv * wr[c];
  }
  float m = l[0];
  #pragma unroll
  for (int c = 1; c < N_CLASSES; ++c) m = fmaxf(m, l[c]);
  float sum = 0.f;
  #pragma unroll
  for (int c = 0; c < N_CLASSES; ++c) sum += expf(l[c] - m);
  float lse = m + logf(sum);
  #pragma unroll
  for (int c = 0; c < N_CLASSES; ++c) logp[g * N_CLASSES + c] = l[c] - lse;
}

// ---------------- host orchestration ----------------
extern "C" void kernel_launch(void* const* d_in, const int* in_sizes, int n_in,
                              void* d_out, int out_size, void* d_ws, size_t ws_size,
                              hipStream_t stream) {
  (void)in_sizes; (void)n_in; (void)out_size; (void)ws_size;

  const float* x    = (const float*)d_in[0];
  const int*   ei   = (const int*)d_in[1];
  /* d_in[2] batch: contiguous segments, not needed */
  const float* W1   = (const float*)d_in[3];
  const float* b1   = (const float*)d_in[4];
  const float* g1   = (const float*)d_in[5];
  const float* be1  = (const float*)d_in[6];
  const float* W2   = (const float*)d_in[7];
  const float* b2   = (const float*)d_in[8];
  const float* g2   = (const float*)d_in[9];
  const float* be2  = (const float*)d_in[10];
  const float* W_ih = (const float*)d_in[11];
  /* d_in[12] W_hh unused: h0 == 0 */
  const float* b_ih = (const float*)d_in[13];
  const float* b_hh = (const float*)d_in[14];
  const float* W_fc = (const float*)d_in[15];
  const float* b_fc = (const float*)d_in[16];

  const int* src = ei;
  const int* dst = ei + N_EDGES;

  float* ws = (float*)d_ws;
  size_t off = 0;
  float* deg    = ws + off; off += N_NODES;
  float* dinv   = ws + off; off += N_NODES;
  float* invdeg = ws + off; off += N_NODES;
  float* colsum = ws + off; off += HID;
  float* colsq  = ws + off; off += HID;     // contiguous with colsum (zeroed together)
  float* scale  = ws + off; off += HID;
  float* shift  = ws + off; off += HID;
  float* pooled = ws + off; off += (size_t)N_GRAPHS * HID;
  float* gates  = ws + off; off += (size_t)N_GRAPHS * 4 * HID;
  float* bufA   = ws + off; off += (size_t)N_NODES * HID;   // linear output  (h = in @ W)
  float* bufB   = ws + off; off += (size_t)N_NODES * HID;   // aggregation / pre-BN
  float* bufC   = ws + off; off += (size_t)N_NODES * HID;   // post-ReLU activations

  float* out_logp = (float*)d_out;
  float* out_h1   = out_logp + N_GRAPHS * N_CLASSES;
  float* out_c1   = out_h1 + N_GRAPHS * HID;

  const size_t NH = (size_t)N_NODES * HID;
  const dim3 gemm_grid_big(N_NODES / 16, HID / 256);        // (2048, 1)
  const dim3 gemm_grid_lstm(N_GRAPHS / 16, (4 * HID) / 256); // (4, 4)
  const int scatter_blocks = (int)(((long long)N_EDGES * 64) / 256);   // 131072

  // degree / normalization
  k_zero<<<128, 256, 0, stream>>>(deg, (size_t)N_NODES);
  k_deg<<<N_EDGES / 256, 256, 0, stream>>>(dst, deg);
  k_dinv<<<N_NODES / 256, 256, 0, stream>>>(deg, dinv, invdeg);

  // ---- GCN layer 1 ----
  k_wmma_gemm256<<<gemm_grid_big, 256, 0, stream>>>(x, W1, bufA, HID);
  k_zero<<<4096, 256, 0, stream>>>(bufB, NH);
  k_scatter<<<scatter_blocks, 256, 0, stream>>>(bufA, src, dst, dinv, bufB);
  k_zero<<<1, 256, 0, stream>>>(colsum, (size_t)(2 * HID));
  k_finalize<<<N_NODES / FIN_ROWS, 256, 0, stream>>>(bufB, bufA, invdeg, b1, colsum, colsq);
  k_bnfin<<<1, 256, 0, stream>>>(colsum, colsq, g1, be1, scale, shift);
  k_bnrelu<<<(int)(NH / 256), 256, 0, stream>>>(bufB, scale, shift, bufC);

  // ---- GCN layer 2 ----
  k_wmma_gemm256<<<gemm_grid_big, 256, 0, stream>>>(bufC, W2, bufA, HID);
  k_zero<<<4096, 256, 0, stream>>>(bufB, NH);
  k_scatter<<<scatter_blocks, 256, 0, stream>>>(bufA, src, dst, dinv, bufB);
  k_zero<<<1, 256, 0, stream>>>(colsum, (size_t)(2 * HID));
  k_finalize<<<N_NODES / FIN_ROWS, 256, 0, stream>>>(bufB, bufA, invdeg, b2, colsum, colsq);
  k_bnfin<<<1, 256, 0, stream>>>(colsum, colsq, g2, be2, scale, shift);
  k_bnrelu<<<(int)(NH / 256), 256, 0, stream>>>(bufB, scale, shift, bufC);

  // ---- pool + LSTM + FC ----
  k_pool<<<N_GRAPHS, 256, 0, stream>>>(bufC, pooled);
  k_wmma_gemm256<<<gemm_grid_lstm, 256, 0, stream>>>(pooled, W_ih, gates, 4 * HID);
  k_lstm<<<(N_GRAPHS * HID) / 256, 256, 0, stream>>>(gates, b_ih, b_hh, out_h1, out_c1);
  k_fc<<<1, 64, 0, stream>>>(out_h1, W_fc, b_fc, out_logp);
}